// GATlayer_6665789243399
// MI455X (gfx1250) — compile-verified
//
#include <hip/hip_runtime.h>

#define N_NODES 50000
#define N_EDGES 800000
#define D_IN    128
#define D_HID   256
#define D_OUT   128

// LDS row stride (floats) for the 16 x 256 hidden tile; 258 % 64 == 2 banks
// -> A-fragment reads (stride-one-row across lanes) spread over 32 banks.
#define H1_STRIDE 258

typedef __attribute__((ext_vector_type(2))) float v2f;
typedef __attribute__((ext_vector_type(8))) float v8f;

// ---------------------------------------------------------------------------
// Fused 2-layer MLP: out16x128 = relu(nh16x128 @ W1 + b1) @ W2 + b2
// One block = 8 wave32 = 16 node rows. WMMA f32 16x16x4, K-stepped by 4.
// A frag: lane(m = l&15) holds row m; VGPR0/1 = K = k0 + (l>>4)*2 (+1).
// B frag: lane holds col (l&15); VGPR0/1 = same K striping (mirror of A).
// C/D:    VGPR v, lane l -> row v + (l>>4)*8, col l&15.
// ---------------------------------------------------------------------------
__global__ __launch_bounds__(256)
void mlp_fused(const float* __restrict__ nh, const float* __restrict__ W1,
               const float* __restrict__ b1, const float* __restrict__ W2,
               const float* __restrict__ b2, float* __restrict__ n_h_ws,
               float* __restrict__ out_nodes) {
    __shared__ float sH1[16 * H1_STRIDE];
    const int r0   = blockIdx.x * 16;
    const int wave = threadIdx.x >> 5;
    const int lane = threadIdx.x & 31;
    const int sub  = lane >> 4;    // 0: K lo pair, 1: K hi pair
    const int m    = lane & 15;    // row (A) / col (B,C,D)

    // ---- Phase 1: hidden tile, each wave does 2 of 16 col-tiles ----
    #pragma unroll
    for (int t = 0; t < 2; ++t) {
        const int n0 = (wave * 2 + t) * 16;
        v8f acc = {};
        for (int k0 = 0; k0 < D_IN; k0 += 4) {
            const int kb = k0 + sub * 2;
            v2f a = *(const v2f*)&nh[(size_t)(r0 + m) * D_IN + kb];
            v2f b;
            b.x = W1[(size_t)kb       * D_HID + n0 + m];
            b.y = W1[(size_t)(kb + 1) * D_HID + n0 + m];
            acc = __builtin_amdgcn_wmma_f32_16x16x4_f32(
                false, a, false, b, (short)0, acc, false, false);
        }
        const float bias = b1[n0 + m];
        #pragma unroll
        for (int v = 0; v < 8; ++v) {
            const int row = v + sub * 8;
            float val = acc[v] + bias;
            sH1[row * H1_STRIDE + n0 + m] = val > 0.f ? val : 0.f;
        }
    }
    __syncthreads();

    // ---- Phase 2: output tile, each wave does 1 of 8 col-tiles, K=256 ----
    {
        const int n0 = wave * 16;
        v8f acc = {};
        for (int k0 = 0; k0 < D_HID; k0 += 4) {
            const int kb = k0 + sub * 2;
            v2f a = *(const v2f*)&sH1[m * H1_STRIDE + kb];   // kb even -> 8B aligned
            v2f b;
            b.x = W2[(size_t)kb       * D_OUT + n0 + m];
            b.y = W2[(size_t)(kb + 1) * D_OUT + n0 + m];
            acc = __builtin_amdgcn_wmma_f32_16x16x4_f32(
                false, a, false, b, (short)0, acc, false, false);
        }
        const float bias = b2[n0 + m];
        #pragma unroll
        for (int v = 0; v < 8; ++v) {
            const int row = v + sub * 8;
            const float val = acc[v] + bias;
            const size_t off = (size_t)(r0 + row) * D_OUT + n0 + m;
            n_h_ws[off]    = val;   // clean copy for edge gathers
            out_nodes[off] = val;   // d_out init = n_h (scatter adds on top)
        }
    }
}

// ---------------------------------------------------------------------------
// Init segment-max (as -inf bit pattern) and denom (0).
// ---------------------------------------------------------------------------
__global__ void seg_init(int* __restrict__ segmax_bits, float* __restrict__ denom, int n) {
    int i = blockIdx.x * blockDim.x + threadIdx.x;
    if (i < n) {
        segmax_bits[i] = 0xFF800000;   // -inf
        denom[i]       = 0.f;
    }
}

// ---------------------------------------------------------------------------
// Wave per edge: attn[e] = dot(n_h[src], n_h[dst]) over 128 dims.
// float4 per lane (b128 loads), wave32 xor-shuffle reduction.
// Float atomic max via signed/unsigned int trick (correct for non-NaN).
// ---------------------------------------------------------------------------
__global__ __launch_bounds__(256)
void edge_dot(const float* __restrict__ n_h, const int* __restrict__ ei,
              float* __restrict__ attn, int* __restrict__ segmax_bits, int nE) {
    const int wid = (blockIdx.x * blockDim.x + threadIdx.x) >> 5;
    if (wid >= nE) return;
    const int lane = threadIdx.x & 31;
    const int src = ei[wid];
    const int dst = ei[nE + wid];
    const float4 a = *(const float4*)&n_h[(size_t)src * D_OUT + lane * 4];
    const float4 b = *(const float4*)&n_h[(size_t)dst * D_OUT + lane * 4];
    float p = a.x * b.x + a.y * b.y + a.z * b.z + a.w * b.w;
    #pragma unroll
    for (int off = 16; off; off >>= 1) p += __shfl_xor(p, off, 32);
    if (lane == 0) {
        attn[wid] = p;
        if (p >= 0.f)
            atomicMax(&segmax_bits[dst], __float_as_int(p));
        else
            atomicMin((unsigned int*)&segmax_bits[dst], __float_as_uint(p));
    }
}

// ---------------------------------------------------------------------------
// Thread per edge: stable exp + denominator accumulation.
// ---------------------------------------------------------------------------
__global__ void edge_exp(const float* __restrict__ attn, const int* __restrict__ ei,
                         const float* __restrict__ segmax, float* __restrict__ ew,
                         float* __restrict__ denom, int nE) {
    int e = blockIdx.x * blockDim.x + threadIdx.x;
    if (e >= nE) return;
    const int dst = ei[nE + e];
    const float x = __expf(attn[e] - segmax[dst]);
    ew[e] = x;
    atomicAdd(&denom[dst], x);
}

// ---------------------------------------------------------------------------
// Wave per edge: out[dst] += (ew/denom) * n_h[src]  (global_atomic_add_f32)
// ---------------------------------------------------------------------------
__global__ __launch_bounds__(256)
void edge_scatter(const float* __restrict__ n_h, const int* __restrict__ ei,
                  const float* __restrict__ ew, const float* __restrict__ denom,
                  float* __restrict__ out_nodes, int nE) {
    const int wid = (blockIdx.x * blockDim.x + threadIdx.x) >> 5;
    if (wid >= nE) return;
    const int lane = threadIdx.x & 31;
    const int src = ei[wid];
    const int dst = ei[nE + wid];
    const float w = ew[wid] / denom[dst];
    const float4 s = *(const float4*)&n_h[(size_t)src * D_OUT + lane * 4];
    float* o = &out_nodes[(size_t)dst * D_OUT + lane * 4];
    atomicAdd(o + 0, w * s.x);
    atomicAdd(o + 1, w * s.y);
    atomicAdd(o + 2, w * s.z);
    atomicAdd(o + 3, w * s.w);
}

extern "C" void kernel_launch(void* const* d_in, const int* in_sizes, int n_in,
                              void* d_out, int out_size, void* d_ws, size_t ws_size,
                              hipStream_t stream) {
    const float* nh = (const float*)d_in[0];
    const float* eh = (const float*)d_in[1];
    const int*   ei = (const int*)d_in[2];
    const float* W1 = (const float*)d_in[3];
    const float* b1 = (const float*)d_in[4];
    const float* W2 = (const float*)d_in[5];
    const float* b2 = (const float*)d_in[6];

    float* out = (float*)d_out;               // [N*128 nodes][E*16 eh]

    // workspace layout (floats)
    float* ws     = (float*)d_ws;
    float* n_h    = ws;                                   // N*128
    float* attn   = n_h  + (size_t)N_NODES * D_OUT;       // E
    float* ew     = attn + N_EDGES;                       // E
    float* segmax = ew   + N_EDGES;                       // N
    float* denom  = segmax + N_NODES;                     // N

    seg_init<<<(N_NODES + 255) / 256, 256, 0, stream>>>((int*)segmax, denom, N_NODES);

    mlp_fused<<<N_NODES / 16, 256, 0, stream>>>(nh, W1, b1, W2, b2, n_h, out);

    const int waves_per_blk = 256 / 32;
    const int edge_blocks   = (N_EDGES + waves_per_blk - 1) / waves_per_blk;

    edge_dot<<<edge_blocks, 256, 0, stream>>>(n_h, ei, attn, (int*)segmax, N_EDGES);
    edge_exp<<<(N_EDGES + 255) / 256, 256, 0, stream>>>(attn, ei, segmax, ew, denom, N_EDGES);
    edge_scatter<<<edge_blocks, 256, 0, stream>>>(n_h, ei, ew, denom, out, N_EDGES);

    // eh pass-through (second output of the tuple)
    hipMemcpyAsync(out + (size_t)N_NODES * D_OUT, eh,
                   (size_t)N_EDGES * 16 * sizeof(float),
                   hipMemcpyDeviceToDevice, stream);
}